// linear_update_11751030522160
// MI455X (gfx1250) — compile-verified
//
#include <hip/hip_runtime.h>
#include <math.h>

// Problem constants from the reference
#define NCc 32
#define NRr 64
#define TT  64
#define DDd 128   // 2*NC + NR

typedef float v4f __attribute__((ext_vector_type(4)));
typedef int   vi4 __attribute__((vector_size(16)));   // matches builtin param type

#ifndef __has_builtin
#define __has_builtin(x) 0
#endif

// ---- CDNA5 async global -> LDS (ASYNCcnt-tracked) -------------------------
__device__ __forceinline__ void async_g2l_b128(const float* g, float* lds_generic) {
#if __has_builtin(__builtin_amdgcn_global_load_async_to_lds_b128)
  __builtin_amdgcn_global_load_async_to_lds_b128(
      (__attribute__((address_space(1))) vi4*)g,
      (__attribute__((address_space(3))) vi4*)lds_generic,
      /*offset=*/0, /*cpol=*/0);
#elif defined(__gfx1250__)
  // GV mode: vdst = LDS byte address (low 32 bits of generic LDS addr),
  // vaddr = 64-bit global address, saddr = off
  unsigned lofs = (unsigned)(unsigned long long)lds_generic;
  asm volatile("global_load_async_to_lds_b128 %0, %1, off"
               :: "v"(lofs), "v"(g) : "memory");
#else
  *(v4f*)lds_generic = *(const v4f*)g;
#endif
}

__device__ __forceinline__ void wait_async0() {
#if __has_builtin(__builtin_amdgcn_s_wait_asynccnt)
  __builtin_amdgcn_s_wait_asynccnt(0);
#else
  asm volatile("s_wait_asynccnt 0" ::: "memory");
#endif
  asm volatile("" ::: "memory");   // keep LDS reads behind the wait
}

// ---------------------------------------------------------------------------
// One wave per batch row. Lane L owns channels [4L, 4L+3].
// Lanes 0..15  -> complex pairs k0=2L, k1=2L+1 (rotation+growth 2x2)
// Lanes 16..31 -> real channels (pure exponential decay, diagonal 2x2)
// Time loop: z_{t+1} = M * z_t  (8 FMAs), NT b128 store per lane per step.
// ---------------------------------------------------------------------------
__global__ __launch_bounds__(256)
void seq_expand_kernel(const float* __restrict__ y,
                       const float* __restrict__ Km,
                       float* __restrict__ out, int B) {
  __shared__ float sm[8][256];   // per wave: [0..127] = y0 row, [128..255] = Km row

  const int tid  = threadIdx.x;
  const int wave = tid >> 5;
  const int lane = tid & 31;
  const int b    = blockIdx.x * 8 + wave;
  if (b >= B) return;

  const int    ch0   = lane * 4;
  const size_t ybase = (size_t)b * TT * DDd;     // y[b,0,:] and out[b,0,:]
  const size_t kbase = (size_t)b * DDd;

  // Stage this wave's y0 row and Km row into LDS via async DMA (2x b128/lane).
  async_g2l_b128(y  + ybase + ch0, &sm[wave][ch0]);
  async_g2l_b128(Km + kbase + ch0, &sm[wave][128 + ch0]);
  wait_async0();   // per-wave private staging: no workgroup barrier needed

  v4f z = *(const v4f*)&sm[wave][ch0];           // state at t=0 (== output t=0)
  const float* kr = &sm[wave][128];              // this row's Km[0..127]

  float A0, B0, C0, D0, A1, B1, C1, D1;
  if (ch0 < 2 * NCc) {
    // complex pairs: step multiplier e^{mu} * (cos th - i sin th)
    const int k0 = ch0 >> 1;                     // = 2*lane
    float e0 = expf(kr[k0]);
    float s0, c0; sincosf(kr[NCc + k0], &s0, &c0);
    A0 = e0 * c0;  B0 = e0 * s0;  C0 = -B0;  D0 = A0;
    const int k1 = k0 + 1;
    float e1 = expf(kr[k1]);
    float s1, c1; sincosf(kr[NCc + k1], &s1, &c1);
    A1 = e1 * c1;  B1 = e1 * s1;  C1 = -B1;  D1 = A1;
  } else {
    // real channels: rho for channel ch lives at Km[ch] (2*NC + (ch-64) == ch)
    A0 = expf(kr[ch0 + 0]);  D0 = expf(kr[ch0 + 1]);  B0 = 0.f;  C0 = 0.f;
    A1 = expf(kr[ch0 + 2]);  D1 = expf(kr[ch0 + 3]);  B1 = 0.f;  C1 = 0.f;
  }

  float* op = out + ybase + ch0;
  #pragma unroll 4
  for (int t = 0; t < TT; ++t) {
    __builtin_nontemporal_store(z, (v4f*)op);    // 256MB stream: bypass caches
    float u0 = fmaf(A0, z.x, B0 * z.y);
    float v0 = fmaf(C0, z.x, D0 * z.y);
    float u1 = fmaf(A1, z.z, B1 * z.w);
    float v1 = fmaf(C1, z.z, D1 * z.w);
    z = (v4f){u0, v0, u1, v1};
    op += DDd;
  }
}

extern "C" void kernel_launch(void* const* d_in, const int* in_sizes, int n_in,
                              void* d_out, int out_size, void* d_ws, size_t ws_size,
                              hipStream_t stream) {
  const float* y  = (const float*)d_in[0];   // (B, T, D) f32 — only t=0 slice read
  const float* Km = (const float*)d_in[1];   // (B, D) f32
  float* out = (float*)d_out;                // (B, T, D) f32
  const int B = in_sizes[1] / DDd;
  const int blocks = (B + 7) / 8;            // 8 rows (waves) per 256-thread block
  seq_expand_kernel<<<blocks, 256, 0, stream>>>(y, Km, out, B);
}